// ChamferDistanceLoss_14233521619089
// MI455X (gfx1250) — compile-verified
//
#include <hip/hip_runtime.h>

// CDNA5 / gfx1250 Chamfer distance via V_WMMA_F32_16X16X4_F32 with B staged
// in LDS in WMMA fragment layout (256 KB per workgroup -- needs CDNA5's
// 320 KB/WG LDS).
//
// d2(i,j) = |a_i|^2 + (|b_j|^2 - 2 a_i.b_j)
// Inner GEMM (K padded 3->4): A[i,:] = (-2ax,-2ay,-2az, 1), B[:,j] = (bx,by,bz,|b_j|^2)
// => (A x B)[i,j] = |b_j|^2 - 2 a_i.b_j.  |a_i|^2 added after the j-min (monotone).
// min_j sqrt(.) == sqrt(min_j .), so sqrt once per row.

typedef __attribute__((ext_vector_type(2))) float v2f;
typedef __attribute__((ext_vector_type(8))) float v8f;

#define WAVES_PER_BLOCK 8
#define BLOCK_THREADS (WAVES_PER_BLOCK * 32)

__global__ __launch_bounds__(BLOCK_THREADS)
void chamfer_wmma_f32_kernel(const float* __restrict__ a,
                             const float* __restrict__ b,
                             float* __restrict__ partials,
                             int n, int m, int ntilesA, int ntilesB) {
    // Dynamic LDS: packed B fragments, ntilesB * 32 lanes * v2f = ntilesB*256 bytes.
    HIP_DYNAMIC_SHARED(float, smem);
    v2f* bpack = (v2f*)smem;

    const int lane = threadIdx.x & 31;
    const bool lo  = (lane < 16);
    const int  rl  = lane & 15;

    // ---- Phase 1: cooperatively pack B into WMMA B-fragment layout ----
    // Entry e = jt*32 + l : lane l of tile jt.
    //   lanes 0-15  -> (bx, by)        (K=0,1)
    //   lanes 16-31 -> (bz, |b|^2)     (K=2,3)
    const int entries = ntilesB * 32;
    for (int e = threadIdx.x; e < entries; e += BLOCK_THREADS) {
        const int  l   = e & 31;
        const int  j   = ((e >> 5) << 4) + (l & 15);
        const bool elo = (l < 16);
        float bx = 0.f, by = 0.f, bz = 0.f;
        float b2 = 3.0e37f;                    // padded cols never win the min
        if (j < m) {
            const float* bp = b + j * 3;
            bx = bp[0]; by = bp[1]; bz = bp[2];
            b2 = fmaf(bx, bx, fmaf(by, by, bz * bz));
        }
        v2f p;
        p.x = elo ? bx : bz;
        p.y = elo ? by : b2;
        bpack[e] = p;                          // ds_store_b64, 8B aligned
    }
    __syncthreads();

    const int it = blockIdx.x * WAVES_PER_BLOCK + (threadIdx.x >> 5); // a-tile
    if (it >= ntilesA) return;                 // wave-uniform (after barrier)

    // ---- Build A fragment (16x4 f32: lanes 0-15 hold K=0,1; lanes 16-31 K=2,3)
    const int row = (it << 4) + rl;
    float ax = 0.f, ay = 0.f, az = 0.f;
    if (row < n) {
        const float* ap = a + row * 3;
        ax = ap[0]; ay = ap[1]; az = ap[2];
    }
    v2f A;
    A.x = lo ? (-2.0f * ax) : (-2.0f * az);    // K=0 : K=2
    A.y = lo ? (-2.0f * ay) : (1.0f);          // K=1 : K=3
    const float a2l = fmaf(ax, ax, fmaf(ay, ay, az * az));

    v8f minv;
#pragma unroll
    for (int r = 0; r < 8; ++r) minv[r] = 3.0e38f;
    const v8f czero = {};

    // ---- Phase 2: sweep all b tiles from LDS: ds_load_b64 + WMMA + 8 mins ----
    const v2f* bl = bpack + lane;              // stride 32 v2f (256 B) per tile
#pragma unroll 4
    for (int jt = 0; jt < ntilesB; ++jt) {
        const v2f B = bl[jt * 32];
        // D[i,j] = |b_j|^2 - 2 a_i.b_j    (C = 0)
        v8f d = __builtin_amdgcn_wmma_f32_16x16x4_f32(
            /*neg_a=*/false, A, /*neg_b=*/false, B,
            /*c_mod=*/(short)0, czero, /*reuse_a=*/false, /*reuse_b=*/false);
#pragma unroll
        for (int r = 0; r < 8; ++r) minv[r] = fminf(minv[r], d[r]);
    }

    // ---- Per-row min across the 16 lanes of each half-wave, +|a|^2, sqrt, sum.
    // D layout: VGPR r holds row M=r (lanes 0-15) and M=r+8 (lanes 16-31).
    float acc = 0.0f;
#pragma unroll
    for (int r = 0; r < 8; ++r) {
        float mv = minv[r];
        mv = fminf(mv, __shfl_xor(mv, 1));
        mv = fminf(mv, __shfl_xor(mv, 2));
        mv = fminf(mv, __shfl_xor(mv, 4));
        mv = fminf(mv, __shfl_xor(mv, 8));     // stays within each 16-lane half
        const float a2lo = __shfl(a2l, r);     // |a|^2 of tile-row r
        const float a2hi = __shfl(a2l, r + 8); // |a|^2 of tile-row r+8
        const float myA2 = lo ? a2lo : a2hi;
        const int myRow  = (it << 4) + (lo ? r : (r + 8));
        float d2   = fmaxf(mv + myA2, 0.0f);
        float dist = __builtin_sqrtf(d2);
        if (myRow < n) acc += dist;            // ignore padded a rows
    }
    // lane 0 carries rows 0-7 of this tile, lane 16 carries rows 8-15.
    const float sumLo = __shfl(acc, 0);
    const float sumHi = __shfl(acc, 16);
    if (lane == 0) partials[it] = sumLo + sumHi;
}

__global__ __launch_bounds__(256)
void chamfer_reduce_kernel(const float* __restrict__ partials,
                           float* __restrict__ out, int cnt) {
    __shared__ float s[256];
    float v = 0.0f;
    for (int i = threadIdx.x; i < cnt; i += 256) v += partials[i];
    s[threadIdx.x] = v;
    __syncthreads();
    for (int off = 128; off > 0; off >>= 1) {
        if ((int)threadIdx.x < off) s[threadIdx.x] += s[threadIdx.x + off];
        __syncthreads();
    }
    if (threadIdx.x == 0) out[0] = s[0];
}

extern "C" void kernel_launch(void* const* d_in, const int* in_sizes, int n_in,
                              void* d_out, int out_size, void* d_ws, size_t ws_size,
                              hipStream_t stream) {
    const float* a = (const float*)d_in[0];
    const float* b = (const float*)d_in[1];
    float* out = (float*)d_out;

    const int n = in_sizes[0] / 3;             // 16384 points
    const int m = in_sizes[1] / 3;             // 16384 points
    const int ntilesA = (n + 15) / 16;         // 1024
    const int ntilesB = (m + 15) / 16;         // 1024

    float* partials = (float*)d_ws;            // ntilesA floats (4 KB)

    const int    blocks   = (ntilesA + WAVES_PER_BLOCK - 1) / WAVES_PER_BLOCK;
    const size_t ldsBytes = (size_t)ntilesB * 32 * sizeof(v2f);  // 256 KB

    chamfer_wmma_f32_kernel<<<blocks, BLOCK_THREADS, ldsBytes, stream>>>(
        a, b, partials, n, m, ntilesA, ntilesB);
    chamfer_reduce_kernel<<<1, 256, 0, stream>>>(partials, out, ntilesA);
}